// Yolo_Detector_28269474742963
// MI455X (gfx1250) — compile-verified
//
#include <hip/hip_runtime.h>
#include <math.h>

#define BATCH   16
#define NDET    22743      // 76*76*3 + 38*38*3 + 19*19*3
#define N76     17328
#define N38     21660
#define NCLS    80
#define KTOP    256
#define CONF_T  0.7f
#define NMS_T   0.6f

typedef __attribute__((ext_vector_type(2))) float v2f;
typedef __attribute__((ext_vector_type(8))) float v8f;

#define AS1I(p) ((__attribute__((address_space(1))) int*)(p))
#define AS3I(p) ((__attribute__((address_space(3))) int*)(p))

__constant__ float c_aw[3][3] = {{28.f,46.f,64.f},{102.f,78.f,132.f},{149.f,174.f,257.f}};
__constant__ float c_ah[3][3] = {{28.f,45.f,66.f},{74.f,115.f,113.f},{163.f,268.f,176.f}};

__device__ __forceinline__ float sigmoidf_(float x) {
  return 1.0f / (1.0f + __expf(-x));
}

// ---------------------------------------------------------------------------
// Kernel A: decode.  One thread per (b, box-row n). Memory-bound (~248 MB
// -> ~11 us at 23.3 TB/s).  out[b][n][0..84]: xy, wh, obj, cls*obj.
// ---------------------------------------------------------------------------
__global__ __launch_bounds__(256) void decode_kernel(
    const float* __restrict__ p76, const float* __restrict__ p38,
    const float* __restrict__ p19, float* __restrict__ out)
{
  int gid = blockIdx.x * 256 + threadIdx.x;
  if (gid >= BATCH * NDET) return;
  int b = gid / NDET;
  int n = gid % NDET;

  const float* p; int S, base, si; float stride;
  if (n < N76)      { p = p76; S = 76; base = 0;    si = 0; stride = 8.f;  }
  else if (n < N38) { p = p38; S = 38; base = N76;  si = 1; stride = 16.f; }
  else              { p = p19; S = 19; base = N38;  si = 2; stride = 32.f; }

  int r  = n - base;
  int hw = r / 3, a = r % 3;
  int h  = hw / S, w = hw % S;

  int cs = S * S;                                   // channel stride
  const float* ip = p + ((size_t)(b * 3 + a) * 85) * cs + (size_t)h * S + w;
  float* op = out + (size_t)gid * 85;

  float t0 = ip[0*(size_t)cs], t1 = ip[1*(size_t)cs];
  float t2 = ip[2*(size_t)cs], t3 = ip[3*(size_t)cs];
  float t4 = ip[4*(size_t)cs];
  float obj = sigmoidf_(t4);

  op[0] = (sigmoidf_(t0) * 1.05f - 0.025f + (float)w) * stride;  // gx = w
  op[1] = (sigmoidf_(t1) * 1.05f - 0.025f + (float)h) * stride;  // gy = h
  op[2] = __expf(t2) * c_aw[si][a];
  op[3] = __expf(t3) * c_ah[si][a];
  op[4] = obj;

  #pragma unroll 4
  for (int c = 5; c < 85; ++c) {
    if (c + 8 < 85) __builtin_prefetch(ip + (size_t)(c + 8) * cs, 0, 0); // global_prefetch_b8
    op[c] = sigmoidf_(ip[(size_t)c * cs]) * obj;
  }
}

// ---------------------------------------------------------------------------
// Kernel B: exact top-K per (b, cls).  Whole group's 22743 scores staged in
// LDS (~91 KB — only possible with CDNA5's 320 KB LDS/WGP).  The strided
// gather (stride 340 B) is done with GLOBAL_LOAD_ASYNC_TO_LDS_B32: no VGPR
// round-trip, one ASYNCcnt drain for the whole stage-in.  Then 256
// deterministic block-argmax extractions with (value desc, index asc)
// tie-break matching jax.lax.top_k.
// ---------------------------------------------------------------------------
#define SPAD 22752   // NDET rounded up for alignment
extern __shared__ float g_lds[];

__global__ __launch_bounds__(256) void topk_kernel(
    const float* __restrict__ out,
    float* __restrict__ vals_ws, unsigned* __restrict__ idx_ws)
{
  int g   = blockIdx.x;            // (b * NCLS + cls)
  int b   = g / NCLS;
  int cls = g % NCLS;
  int t   = threadIdx.x;

  float*    s  = g_lds;                       // [NDET] scores
  float*    rv = g_lds + SPAD;                // [256] reduction values
  unsigned* ri = (unsigned*)(rv + 256);       // [256] reduction indices

  const float* src = out + (size_t)b * NDET * 85 + 5 + cls;

#if __has_builtin(__builtin_amdgcn_global_load_async_to_lds_b32)
  for (int n = t; n < NDET; n += 256) {
    __builtin_amdgcn_global_load_async_to_lds_b32(
        AS1I(src + (size_t)n * 85), AS3I(s + n), 0, 0);
  }
  #if __has_builtin(__builtin_amdgcn_s_wait_asynccnt)
  __builtin_amdgcn_s_wait_asynccnt(0);
  #else
  asm volatile("s_wait_asynccnt 0x0" ::: "memory");
  #endif
#else
  for (int n = t; n < NDET; n += 256) s[n] = src[(size_t)n * 85];
#endif
  __syncthreads();

  for (int k = 0; k < KTOP; ++k) {
    float    bv = -1.0f;            // scores are in (0,1); -1 / -2 are sentinels
    unsigned bi = 0xffffffffu;
    for (int n = t; n < NDET; n += 256) {
      float v = s[n];
      if (v > bv) { bv = v; bi = (unsigned)n; }     // ascending n -> ties keep min n
    }
    rv[t] = bv; ri[t] = bi;
    __syncthreads();
    for (int off = 128; off > 0; off >>= 1) {
      if (t < off) {
        float v1 = rv[t], v2 = rv[t + off];
        unsigned i1 = ri[t], i2 = ri[t + off];
        if (v2 > v1 || (v2 == v1 && i2 < i1)) { rv[t] = v2; ri[t] = i2; }
      }
      __syncthreads();
    }
    if (t == 0) {
      float wv = rv[0]; unsigned wi = ri[0];
      vals_ws[(size_t)g * KTOP + k] = wv;
      idx_ws [(size_t)g * KTOP + k] = wi;
      if (wi < NDET) s[wi] = -2.0f;              // remove winner
    }
    __syncthreads();
  }
}

// ---------------------------------------------------------------------------
// Kernel C: DIoU-NMS + det output.  One block per (b, cls), thread j owns
// box j.  Pairwise r2 = |c_i - c_j|^2 on the matrix pipe:
//   r2 = A x B,  A row i = (cx_i, cy_i, sq_i, 1), B col j = (-2cx_j, -2cy_j, 1, sq_j)
// via V_WMMA_F32_16X16X4_F32, one 16x256 panel per 16 scan steps (16 tiles
// over the block's 8 wave32s), overlapping XDL with the scan's VALU work.
// ---------------------------------------------------------------------------
#define PANW 257   // padded panel row stride (avoid LDS bank conflicts)

__global__ __launch_bounds__(256) void nms_kernel(
    const float* __restrict__ out,
    const float* __restrict__ vals_ws, const unsigned* __restrict__ idx_ws,
    float* __restrict__ det)
{
  __shared__ float sx1[KTOP], sy1[KTOP], sx2[KTOP], sy2[KTOP], sarea[KTOP];
  __shared__ float scx[KTOP], scy[KTOP], ssq[KTOP];
  __shared__ int   sact[KTOP];
  __shared__ float panel[16 * PANW];

  int g   = blockIdx.x;
  int b   = g / NCLS;
  int cls = g % NCLS;
  int t   = threadIdx.x;

  unsigned id  = idx_ws[(size_t)g * KTOP + t];
  float    val = vals_ws[(size_t)g * KTOP + t];

  const float* brow = out + ((size_t)b * NDET + id) * 85;
  float cx = brow[0], cy = brow[1], w = brow[2], h = brow[3];

  float x1 = cx - 0.5f * w, y1 = cy - 0.5f * h;
  float x2 = cx + 0.5f * w, y2 = cy + 0.5f * h;
  float area = w * h;

  scx[t] = cx; scy[t] = cy; ssq[t] = cx * cx + cy * cy;
  sx1[t] = x1; sy1[t] = y1; sx2[t] = x2; sy2[t] = y2; sarea[t] = area;
  int active = (val > CONF_T) ? 1 : 0;
  sact[t] = active;
  __syncthreads();

  int lane = t & 31, wid = t >> 5;
  int m16 = lane & 15, hi = lane >> 4;
  int keep = 0;

  for (int p = 0; p < 16; ++p) {
    // --- WMMA: r2 panel for scan rows [p*16, p*16+16) ------------------
    {
      int i = p * 16 + m16;
      v2f A;
      if (hi) { A.x = ssq[i]; A.y = 1.0f; }          // lanes 16-31: K=2,3
      else    { A.x = scx[i]; A.y = scy[i]; }        // lanes  0-15: K=0,1
      #pragma unroll
      for (int tt = 0; tt < 2; ++tt) {
        int tj = wid * 2 + tt;                       // 2 of the 16 column-tiles
        int j  = tj * 16 + m16;
        v2f Bv;
        if (hi) { Bv.x = 1.0f;            Bv.y = ssq[j]; }          // K=2,3
        else    { Bv.x = -2.0f * scx[j];  Bv.y = -2.0f * scy[j]; }  // K=0,1
        v8f C = {};
        C = __builtin_amdgcn_wmma_f32_16x16x4_f32(
                false, A, false, Bv, (short)0, C, false, false);
        #pragma unroll
        for (int vv = 0; vv < 8; ++vv) {
          int mrow = vv + (hi ? 8 : 0);              // D: M = v + 8*(lane>=16)
          panel[mrow * PANW + tj * 16 + m16] = C[vv];
        }
      }
    }
    __syncthreads();

    // --- sequential suppression scan over this panel -------------------
    for (int il = 0; il < 16; ++il) {
      int i = p * 16 + il;
      int act_i = sact[i];
      if (t == i) keep = act_i;                      // keep[i] = active at step i
      if (act_i) {
        float ix1 = sx1[i], iy1 = sy1[i], ix2 = sx2[i], iy2 = sy2[i];
        float iw = fmaxf(fminf(ix2, x2) - fmaxf(ix1, x1), 0.f);
        float ih = fmaxf(fminf(iy2, y2) - fmaxf(iy1, y1), 0.f);
        float ai = iw * ih;
        float au = sarea[i] + area - ai;
        float ow = fmaxf(fmaxf(ix2, x2) - fminf(ix1, x1), 0.f);
        float oh = fmaxf(fmaxf(iy2, y2) - fminf(iy1, y1), 0.f);
        float c2 = ow * ow + oh * oh;
        float r2 = panel[il * PANW + t];
        float diou = ai / au - r2 / c2;
        if (diou >= NMS_T) active = 0;               // includes j == i (diou ~ 1)
      }
      __syncthreads();
      sact[t] = active;
      __syncthreads();
    }
  }

  // --- det = concat(cls_id, boxes, vals) * keep -------------------------
  float kf = keep ? 1.0f : 0.0f;
  float* drow = det + ((size_t)g * KTOP + t) * 6;
  drow[0] = kf * (float)cls;
  drow[1] = kf * cx;
  drow[2] = kf * cy;
  drow[3] = kf * w;
  drow[4] = kf * h;
  drow[5] = kf * val;
}

// ---------------------------------------------------------------------------
extern "C" void kernel_launch(void* const* d_in, const int* in_sizes, int n_in,
                              void* d_out, int out_size, void* d_ws, size_t ws_size,
                              hipStream_t stream) {
  (void)in_sizes; (void)n_in; (void)out_size; (void)ws_size;
  const float* p76 = (const float*)d_in[0];
  const float* p38 = (const float*)d_in[1];
  const float* p19 = (const float*)d_in[2];

  float* out = (float*)d_out;                           // (16, 22743, 85)
  float* det = out + (size_t)BATCH * NDET * 85;         // (16, 80, 256, 6)

  float*    vals_ws = (float*)d_ws;                                       // 1280*256 f32
  unsigned* idx_ws  = (unsigned*)((char*)d_ws +
                       (size_t)BATCH * NCLS * KTOP * sizeof(float));      // 1280*256 u32

  int total = BATCH * NDET;
  decode_kernel<<<(total + 255) / 256, 256, 0, stream>>>(p76, p38, p19, out);

  size_t ldsB = (size_t)(SPAD + 256) * sizeof(float) + 256 * sizeof(unsigned);
  topk_kernel<<<BATCH * NCLS, 256, ldsB, stream>>>(out, vals_ws, idx_ws);

  nms_kernel<<<BATCH * NCLS, 256, 0, stream>>>(out, vals_ws, idx_ws, det);
}